// MixupGenerator_22720376996225
// MI455X (gfx1250) — compile-verified
//
#include <hip/hip_runtime.h>
#include <hip/hip_bf16.h>
#include <math.h>

// ---------------------------------------------------------------------------
// Shapes: B=4096, K=16, F=512, H=64, C=1000
// d_out = [ z_aug (B*F floats) | y_aug (B*C floats) ]
// ---------------------------------------------------------------------------
#define Bn 4096
#define Kn 16
#define Fn 512
#define Hn 64
#define Cn 1000

typedef __attribute__((ext_vector_type(2))) float v2f;
typedef __attribute__((ext_vector_type(8))) float v8f;
typedef __attribute__((ext_vector_type(4))) int   v4i;

// ---------------------------------------------------------------------------
// Async global -> LDS fill of a 32KB tile (8192 floats) by 512 threads.
// Uses GLOBAL_LOAD_ASYNC_TO_LDS_B128 (ASYNCcnt) when available; otherwise a
// plain float4 copy. Caller must __syncthreads() after this returns.
// ---------------------------------------------------------------------------
__device__ __forceinline__ void fill_lds_32k(float* lds_dst, const float* gsrc, int tid) {
#if defined(__AMDGCN__) && __has_builtin(__builtin_amdgcn_global_load_async_to_lds_b128)
    typedef v4i __attribute__((address_space(1)))* gvec_t;   // global int4*
    typedef v4i __attribute__((address_space(3)))* lvec_t;   // LDS int4*
    #pragma unroll
    for (int s = 0; s < 4; s++) {
        int off = s * 2048 + tid * 4;                 // float index, 16B/lane
        __builtin_amdgcn_global_load_async_to_lds_b128(
            (gvec_t)(gsrc + off), (lvec_t)(lds_dst + off), 0, 0);
    }
#if __has_builtin(__builtin_amdgcn_s_wait_asynccnt)
    __builtin_amdgcn_s_wait_asynccnt(0);
#else
    asm volatile("s_wait_asynccnt 0x0" ::: "memory");
#endif
#else
    #pragma unroll
    for (int s = 0; s < 4; s++) {
        int off = s * 2048 + tid * 4;
        *(float4*)(lds_dst + off) = *(const float4*)(gsrc + off);
    }
#endif
}

// ---------------------------------------------------------------------------
// WV (512x512) -> WVT (transpose), so gemm B operand is row-major/coalesced.
// ---------------------------------------------------------------------------
__global__ void transpose512(const float* __restrict__ in, float* __restrict__ out) {
    __shared__ float tile[32][33];
    int bx = blockIdx.x * 32, by = blockIdx.y * 32;
    int tx = threadIdx.x & 31, ty = threadIdx.x >> 5;   // 256 threads: 32 x 8
    #pragma unroll
    for (int r = 0; r < 32; r += 8)
        tile[ty + r][tx] = in[(size_t)(by + ty + r) * Fn + bx + tx];
    __syncthreads();
    #pragma unroll
    for (int r = 0; r < 32; r += 8)
        out[(size_t)(bx + ty + r) * Fn + by + tx] = tile[tx][ty + r];
}

// ---------------------------------------------------------------------------
// WMMA f32 16x16x4 GEMM.  C(MxN) = A(MxK) * B(KxN), all row-major.
// block = 256 threads = 8 waves; wave tile 16(M) x 64(N); block tile 128 x 64.
// A lane layout (16x4 f32): lanes 0-15 -> K={0,1}, lanes 16-31 -> K={2,3},
// row M = lane&15.  B (4x16): same K split, col N = lane&15.
// D (16x16 f32, 8 VGPRs): VGPR i -> row i (lanes 0-15) / row i+8 (lanes 16-31).
// ---------------------------------------------------------------------------
__global__ void __launch_bounds__(256)
gemm_wmma_nn(const float* __restrict__ A, const float* __restrict__ B,
             float* __restrict__ C, int M, int N, int Kd) {
    const int wave = threadIdx.x >> 5, lane = threadIdx.x & 31;
    const int m0 = blockIdx.x * 128 + wave * 16;
    const int n0 = blockIdx.y * 64;
    const int half = lane >> 4, r = lane & 15;

    v8f acc0 = {}, acc1 = {}, acc2 = {}, acc3 = {};
    const float* arow = A + (size_t)(m0 + r) * Kd + 2 * half; // contiguous pair
    #pragma unroll 4
    for (int k = 0; k < Kd; k += 4) {
        v2f a; a.x = arow[k]; a.y = arow[k + 1];
        const float* bp = B + (size_t)(k + 2 * half) * N + n0 + r;
        v2f b0, b1, b2, b3;
        b0.x = bp[0];  b0.y = bp[N];
        b1.x = bp[16]; b1.y = bp[N + 16];
        b2.x = bp[32]; b2.y = bp[N + 32];
        b3.x = bp[48]; b3.y = bp[N + 48];
        acc0 = __builtin_amdgcn_wmma_f32_16x16x4_f32(false, a, false, b0, (short)0, acc0, false, false);
        acc1 = __builtin_amdgcn_wmma_f32_16x16x4_f32(false, a, false, b1, (short)0, acc1, false, false);
        acc2 = __builtin_amdgcn_wmma_f32_16x16x4_f32(false, a, false, b2, (short)0, acc2, false, false);
        acc3 = __builtin_amdgcn_wmma_f32_16x16x4_f32(false, a, false, b3, (short)0, acc3, false, false);
    }
    float* crow = C + (size_t)(m0 + 8 * half) * N + n0 + r;
    #pragma unroll
    for (int i = 0; i < 8; i++) {
        crow[(size_t)i * N + 0]  = acc0[i];
        crow[(size_t)i * N + 16] = acc1[i];
        crow[(size_t)i * N + 32] = acc2[i];
        crow[(size_t)i * N + 48] = acc3[i];
    }
}

// C(MxN) = A^T * B where A stored (Kd x M) row-major (used for G = WQ^T WQ).
__global__ void __launch_bounds__(256)
gemm_wmma_tn(const float* __restrict__ A, const float* __restrict__ B,
             float* __restrict__ C, int M, int N, int Kd) {
    const int wave = threadIdx.x >> 5, lane = threadIdx.x & 31;
    const int m0 = blockIdx.x * 128 + wave * 16;
    const int n0 = blockIdx.y * 64;
    const int half = lane >> 4, r = lane & 15;

    v8f acc0 = {}, acc1 = {}, acc2 = {}, acc3 = {};
    #pragma unroll 4
    for (int k = 0; k < Kd; k += 4) {
        v2f a;
        a.x = A[(size_t)(k + 2 * half)     * M + m0 + r];   // coalesced over r
        a.y = A[(size_t)(k + 2 * half + 1) * M + m0 + r];
        const float* bp = B + (size_t)(k + 2 * half) * N + n0 + r;
        v2f b0, b1, b2, b3;
        b0.x = bp[0];  b0.y = bp[N];
        b1.x = bp[16]; b1.y = bp[N + 16];
        b2.x = bp[32]; b2.y = bp[N + 32];
        b3.x = bp[48]; b3.y = bp[N + 48];
        acc0 = __builtin_amdgcn_wmma_f32_16x16x4_f32(false, a, false, b0, (short)0, acc0, false, false);
        acc1 = __builtin_amdgcn_wmma_f32_16x16x4_f32(false, a, false, b1, (short)0, acc1, false, false);
        acc2 = __builtin_amdgcn_wmma_f32_16x16x4_f32(false, a, false, b2, (short)0, acc2, false, false);
        acc3 = __builtin_amdgcn_wmma_f32_16x16x4_f32(false, a, false, b3, (short)0, acc3, false, false);
    }
    float* crow = C + (size_t)(m0 + 8 * half) * N + n0 + r;
    #pragma unroll
    for (int i = 0; i < 8; i++) {
        crow[(size_t)i * N + 0]  = acc0[i];
        crow[(size_t)i * N + 16] = acc1[i];
        crow[(size_t)i * N + 32] = acc2[i];
        crow[(size_t)i * N + 48] = acc3[i];
    }
}

// ---------------------------------------------------------------------------
// Per-batch: entropy -> MLP -> softmax(eps) -> z_eps.   1 block / b, 512 thr.
// Z[b] (16x512 = 32KB) staged in LDS via async-to-LDS; wave w reduces row k=w.
// ---------------------------------------------------------------------------
__global__ void __launch_bounds__(512)
prep_kernel(const float* __restrict__ Z, const float* __restrict__ W1,
            const float* __restrict__ b1, const float* __restrict__ W2,
            const float* __restrict__ b2, float* __restrict__ zeps_out) {
    __shared__ __align__(16) float zs[Kn * Fn];
    __shared__ float red[512];
    __shared__ float ent[Kn];
    __shared__ float hbuf[Hn];
    __shared__ float eps[Kn];

    const int b = blockIdx.x, tid = threadIdx.x;
    const float* Zb = Z + (size_t)b * Kn * Fn;
    fill_lds_32k(zs, Zb, tid);
    __syncthreads();

    const int w = tid >> 5, lane = tid & 31;
    float s = 0.f;
    #pragma unroll
    for (int j = 0; j < 16; j++) {
        float p = zs[w * Fn + lane + 32 * j];
        p = fminf(fmaxf(p, 1e-8f), 1.0f - 1e-8f);
        s += p * __logf(p);                      // v_log_f32 path
    }
    red[tid] = s; __syncthreads();
    if (lane < 16) red[tid] += red[tid + 16]; __syncthreads();
    if (lane < 8)  red[tid] += red[tid + 8];  __syncthreads();
    if (lane < 4)  red[tid] += red[tid + 4];  __syncthreads();
    if (lane < 2)  red[tid] += red[tid + 2];  __syncthreads();
    if (lane == 0) ent[w] = -(red[tid] + red[tid + 1]);
    __syncthreads();

    if (tid < Hn) {                               // h = relu(ent @ W1^T + b1)
        float acc = b1[tid];
        #pragma unroll
        for (int k = 0; k < Kn; k++) acc += W1[tid * Kn + k] * ent[k];
        hbuf[tid] = fmaxf(acc, 0.f);
    }
    __syncthreads();
    if (tid < Kn) {                               // logits = h @ W2^T + b2
        float acc = b2[tid];
        #pragma unroll
        for (int j = 0; j < Hn; j++) acc += W2[tid * Hn + j] * hbuf[j];
        red[tid] = acc;
    }
    __syncthreads();
    if (tid == 0) {                               // softmax over K=16
        float mx = red[0];
        for (int k = 1; k < Kn; k++) mx = fmaxf(mx, red[k]);
        float sm = 0.f, e[Kn];
        for (int k = 0; k < Kn; k++) { e[k] = __expf(red[k] - mx); sm += e[k]; }
        float inv = 1.f / sm;
        for (int k = 0; k < Kn; k++) eps[k] = e[k] * inv;
    }
    __syncthreads();

    float acc = 0.f;                              // z_eps[f] = sum_k eps[k]*Z[k,f]
    #pragma unroll
    for (int k = 0; k < Kn; k++) acc += eps[k] * zs[k * Fn + tid];
    zeps_out[(size_t)b * Fn + tid] = acc;
}

// ---------------------------------------------------------------------------
// Per-batch: scores = (u . Z[b,k])/sqrt(F) -> softmax A -> zbar = A @ Z[b].
// ---------------------------------------------------------------------------
__global__ void __launch_bounds__(512)
attn_kernel(const float* __restrict__ Z, const float* __restrict__ u,
            float* __restrict__ zbar_out, float* __restrict__ A_out) {
    __shared__ __align__(16) float zs[Kn * Fn];
    __shared__ float uv[Fn];
    __shared__ float red[512];
    __shared__ float sc[Kn];
    __shared__ float Aw[Kn];

    const int b = blockIdx.x, tid = threadIdx.x;
    const float* Zb = Z + (size_t)b * Kn * Fn;
    uv[tid] = u[(size_t)b * Fn + tid];
    fill_lds_32k(zs, Zb, tid);
    __syncthreads();

    const int w = tid >> 5, lane = tid & 31;
    float s = 0.f;
    #pragma unroll
    for (int j = 0; j < 16; j++) {
        int f = lane + 32 * j;
        s += uv[f] * zs[w * Fn + f];
    }
    red[tid] = s; __syncthreads();
    if (lane < 16) red[tid] += red[tid + 16]; __syncthreads();
    if (lane < 8)  red[tid] += red[tid + 8];  __syncthreads();
    if (lane < 4)  red[tid] += red[tid + 4];  __syncthreads();
    if (lane < 2)  red[tid] += red[tid + 2];  __syncthreads();
    if (lane == 0) sc[w] = (red[tid] + red[tid + 1]) * 0.044194173824159216f; // 1/sqrt(512)
    __syncthreads();

    if (tid == 0) {
        float mx = sc[0];
        for (int k = 1; k < Kn; k++) mx = fmaxf(mx, sc[k]);
        float sm = 0.f, e[Kn];
        for (int k = 0; k < Kn; k++) { e[k] = __expf(sc[k] - mx); sm += e[k]; }
        float inv = 1.f / sm;
        for (int k = 0; k < Kn; k++) Aw[k] = e[k] * inv;
    }
    __syncthreads();
    if (tid < Kn) A_out[(size_t)b * Kn + tid] = Aw[tid];

    float acc = 0.f;
    #pragma unroll
    for (int k = 0; k < Kn; k++) acc += Aw[k] * zs[k * Fn + tid];
    zbar_out[(size_t)b * Fn + tid] = acc;
}

// ---------------------------------------------------------------------------
// y_aug[b,c] = sum_k A[b,k] * y_lb[b,k,c]   (streams 262MB; float4 / b128)
// Row stride 1000 floats = 4000B (16B aligned), so float4 views are legal.
// ---------------------------------------------------------------------------
__global__ void __launch_bounds__(256)
yaug_kernel(const float* __restrict__ ylb, const float* __restrict__ A_in,
            float* __restrict__ yout) {
    __shared__ float Aw[Kn];
    const int b = blockIdx.x, tid = threadIdx.x;
    if (tid < Kn) Aw[tid] = A_in[(size_t)b * Kn + tid];
    __syncthreads();
    if (tid < Cn / 4) {                                    // 250 float4 columns
        const float4* Yb4 = (const float4*)(ylb + (size_t)b * Kn * Cn);
        float4 acc = make_float4(0.f, 0.f, 0.f, 0.f);
        #pragma unroll
        for (int k = 0; k < Kn; k++) {
            float4 v = Yb4[k * (Cn / 4) + tid];            // global_load_b128
            acc.x += Aw[k] * v.x;
            acc.y += Aw[k] * v.y;
            acc.z += Aw[k] * v.z;
            acc.w += Aw[k] * v.w;
        }
        ((float4*)(yout + (size_t)b * Cn))[tid] = acc;     // global_store_b128
    }
}

// ---------------------------------------------------------------------------
extern "C" void kernel_launch(void* const* d_in, const int* in_sizes, int n_in,
                              void* d_out, int out_size, void* d_ws, size_t ws_size,
                              hipStream_t stream) {
    const float* Z   = (const float*)d_in[0];
    const float* ylb = (const float*)d_in[1];
    const float* W1  = (const float*)d_in[2];
    const float* b1  = (const float*)d_in[3];
    const float* W2  = (const float*)d_in[4];
    const float* b2  = (const float*)d_in[5];
    const float* WQ  = (const float*)d_in[6];
    const float* WV  = (const float*)d_in[7];

    float* zaug = (float*)d_out;                    // (B, F)
    float* yaug = zaug + (size_t)Bn * Fn;           // (B, C)

    char* ws = (char*)d_ws;
    float* G    = (float*)(ws + 0);                 // 512*512        = 1 MB
    float* WVT  = (float*)(ws + (size_t)(1  << 20));// 512*512        = 1 MB
    float* zeps = (float*)(ws + (size_t)(2  << 20));// 4096*512       = 8 MB
    float* uu   = (float*)(ws + (size_t)(10 << 20));// 4096*512       = 8 MB
    float* zbar = (float*)(ws + (size_t)(18 << 20));// 4096*512       = 8 MB
    float* Aw   = (float*)(ws + (size_t)(26 << 20));// 4096*16        = 256 KB

    // WVT = WV^T
    transpose512<<<dim3(16, 16), 256, 0, stream>>>(WV, WVT);
    // G = WQ^T * WQ     (512x512x1536)
    gemm_wmma_tn<<<dim3(4, 8), 256, 0, stream>>>(WQ, WQ, G, 512, 512, 1536);
    // entropy -> eps -> z_eps
    prep_kernel<<<Bn, 512, 0, stream>>>(Z, W1, b1, W2, b2, zeps);
    // u = z_eps @ G     (4096x512x512)
    gemm_wmma_nn<<<dim3(32, 8), 256, 0, stream>>>(zeps, G, uu, Bn, Fn, Fn);
    // scores/softmax -> A, zbar
    attn_kernel<<<Bn, 512, 0, stream>>>(Z, uu, zbar, Aw);
    // z_aug = zbar @ WV^T   (4096x512x512)
    gemm_wmma_nn<<<dim3(32, 8), 256, 0, stream>>>(zbar, WVT, zaug, Bn, Fn, Fn);
    // y_aug = A @ y_lb
    yaug_kernel<<<Bn, 256, 0, stream>>>(ylb, Aw, yaug);
}